// GCN_4_layer_edge_weight_fc2_45311904973176
// MI455X (gfx1250) — compile-verified
//
#include <hip/hip_runtime.h>

#define N_NODES 50000
#define N_EDGES 800000
#define DIM     128
#define NCLS    16

typedef __attribute__((ext_vector_type(2))) float v2f;
typedef __attribute__((ext_vector_type(8))) float v8f;

// ---------------------------------------------------------------- utilities
__global__ void zero_f32(float* __restrict__ p, int n) {
    int i = blockIdx.x * blockDim.x + threadIdx.x;
    if (i < n) p[i] = 0.0f;
}

__global__ void degree_accum(const int* __restrict__ src, const int* __restrict__ dst,
                             float* __restrict__ dout, float* __restrict__ din) {
    int e = blockIdx.x * blockDim.x + threadIdx.x;
    if (e < N_EDGES) {
        atomicAdd(&dout[src[e]], 1.0f);
        atomicAdd(&din[dst[e]], 1.0f);
    }
}

__global__ void degree_finalize(float* __restrict__ dout, float* __restrict__ din) {
    int i = blockIdx.x * blockDim.x + threadIdx.x;
    if (i < N_NODES) {
        dout[i] = rsqrtf(fmaxf(dout[i], 1.0f));
        din[i]  = rsqrtf(fmaxf(din[i], 1.0f));
    }
}

// ---------------------------------------------------- WMMA fp32 GEMM (K=128)
// Out[M x NOUT] = (rowscale ? X*rowscale[:,None] : X) @ W (+bias) (opt relu)
// Block: MROWS-row strip of A in LDS (scaled on load). Each wave owns one
// 16-col tile and MROWS/16 row tiles, so every B v2f feeds MROWS/16 WMMAs.
// NOUT is compile-time so all B loads are immediate-offset global_load_b32
// from one per-lane base pointer (no per-step 64-bit address math).
template <int NOUT, int MROWS>
__global__ void __launch_bounds__(256)
gemm_wmma_f32(const float* __restrict__ X,        // M x 128
              const float* __restrict__ W,        // 128 x NOUT, row-major
              const float* __restrict__ rowscale, // nullable, len M
              const float* __restrict__ bias,     // nullable, len NOUT
              float* __restrict__ Out,            // M x NOUT
              int relu) {
    constexpr int MT = MROWS / 16;                // row tiles per wave
    __shared__ float As[MROWS * DIM];
    const int row0 = blockIdx.x * MROWS;

    // cooperative A load (float4), fused row scaling
    {
        const float4* X4  = (const float4*)(X + (size_t)row0 * DIM);
        float4*       As4 = (float4*)As;
        for (int i = threadIdx.x; i < MROWS * (DIM / 4); i += blockDim.x) {
            int r = i >> 5;                       // DIM/4 = 32 float4 per row
            float s = rowscale ? rowscale[row0 + r] : 1.0f;
            float4 v = X4[i];
            v.x *= s; v.y *= s; v.z *= s; v.w *= s;
            As4[i] = v;
        }
    }
    __syncthreads();

    const int wave = threadIdx.x >> 5;
    const int lane = threadIdx.x & 31;
    const int n0   = wave * 16;
    if (n0 >= NOUT) return;
    const int m    = lane & 15;                   // row within tile / col within tile
    const int half = lane >> 4;                   // selects K pair {0,1} vs {2,3}

    // B base: lane covers column n0+m; rows ka = kb + 2*half and ka+1
    const float* wp = W + (2 * half) * NOUT + n0 + m;

    v8f acc[MT];
#pragma unroll
    for (int t = 0; t < MT; ++t) acc[t] = (v8f){};

#pragma unroll
    for (int kb = 0; kb < DIM; kb += 4) {
        v2f b;
        b.x = wp[kb * NOUT];                      // immediate offsets
        b.y = wp[kb * NOUT + NOUT];
#pragma unroll
        for (int t = 0; t < MT; ++t) {
            v2f a;
            a.x = As[(t * 16 + m) * DIM + kb + 2 * half];
            a.y = As[(t * 16 + m) * DIM + kb + 2 * half + 1];
            acc[t] = __builtin_amdgcn_wmma_f32_16x16x4_f32(
                false, a, false, b, (short)0, acc[t], false, false);
        }
    }

    // C/D: VGPR r holds M=r (lanes 0-15) / M=8+r (lanes 16-31), N=lane%16
    const int col = n0 + m;
    const float bv = bias ? bias[col] : 0.0f;
#pragma unroll
    for (int t = 0; t < MT; ++t) {
#pragma unroll
        for (int r = 0; r < 8; ++r) {
            int row = row0 + t * 16 + r + 8 * half;
            float v = acc[t][r] + bv;
            if (relu) v = fmaxf(v, 0.0f);
            Out[(size_t)row * NOUT + col] = v;
        }
    }
}

// --------------------------------------------------- edge gather/scale/scatter
// One wave per edge; lane handles 4 contiguous features (32*4 = 128).
// src/dst/ew are wave-uniform -> scalar loads; agg updated with f32 atomics
// (no return). h and agg (25.6 MB each) are L2-resident on the 192 MB L2.
__global__ void __launch_bounds__(256)
edge_scatter(const int* __restrict__ src, const int* __restrict__ dst,
             const float* __restrict__ ew, const float* __restrict__ h,
             float* __restrict__ agg) {
    int e = blockIdx.x * (blockDim.x >> 5) + (threadIdx.x >> 5);
    if (e >= N_EDGES) return;
    const int lane = threadIdx.x & 31;
    const int s = src[e];
    const int d = dst[e];
    const float w = ew[e];
    const float4* hp = (const float4*)(h + (size_t)s * DIM);
    float4 v = hp[lane];
    float* ap = agg + (size_t)d * DIM + lane * 4;
    atomicAdd(ap + 0, v.x * w);
    atomicAdd(ap + 1, v.y * w);
    atomicAdd(ap + 2, v.z * w);
    atomicAdd(ap + 3, v.w * w);
}

// ------------------------------------------------------------- conv epilogue
// out = agg * isi[row] + bias[col] (+res) ; optional relu  (float4-wide)
__global__ void conv_epilogue(const float4* __restrict__ agg, const float* __restrict__ isi,
                              const float4* __restrict__ bias, const float4* __restrict__ res,
                              float4* __restrict__ out, int relu) {
    int i = blockIdx.x * blockDim.x + threadIdx.x;
    if (i >= N_NODES * (DIM / 4)) return;
    int row = i >> 5;                 // 32 float4 per row
    int col4 = i & 31;
    float s = isi[row];
    float4 a = agg[i];
    float4 b = bias[col4];
    float4 v = {a.x * s + b.x, a.y * s + b.y, a.z * s + b.z, a.w * s + b.w};
    if (res) {
        float4 rr = res[i];
        v.x += rr.x; v.y += rr.y; v.z += rr.z; v.w += rr.w;
    }
    if (relu) {
        v.x = fmaxf(v.x, 0.0f); v.y = fmaxf(v.y, 0.0f);
        v.z = fmaxf(v.z, 0.0f); v.w = fmaxf(v.w, 0.0f);
    }
    out[i] = v;
}

// ------------------------------------------------------------------- driver
extern "C" void kernel_launch(void* const* d_in, const int* in_sizes, int n_in,
                              void* d_out, int out_size, void* d_ws, size_t ws_size,
                              hipStream_t stream) {
    const int*   src    = (const int*)d_in[0];
    const int*   dst    = (const int*)d_in[1];
    const float* inputs = (const float*)d_in[2];
    const float* ew     = (const float*)d_in[3];
    const float* W1 = (const float*)d_in[4];  const float* b1 = (const float*)d_in[5];
    const float* W2 = (const float*)d_in[6];  const float* b2 = (const float*)d_in[7];
    const float* W3 = (const float*)d_in[8];  const float* b3 = (const float*)d_in[9];
    const float* W4 = (const float*)d_in[10]; const float* b4 = (const float*)d_in[11];
    const float* Wr = (const float*)d_in[12]; const float* br = (const float*)d_in[13];
    const float* Wo = (const float*)d_in[14]; const float* bo = (const float*)d_in[15];

    // workspace layout (fp32): iso[N] isi[N] x[N*D] h[N*D] agg[N*D] res[N*D]
    float* ws   = (float*)d_ws;
    float* iso  = ws;
    float* isi  = iso  + N_NODES;
    float* bufx = isi  + N_NODES;
    float* bufh = bufx + (size_t)N_NODES * DIM;
    float* bufa = bufh + (size_t)N_NODES * DIM;
    float* bufr = bufa + (size_t)N_NODES * DIM;

    // 1) degree -> rsqrt scales
    zero_f32<<<(2 * N_NODES + 255) / 256, 256, 0, stream>>>(iso, 2 * N_NODES);
    degree_accum<<<(N_EDGES + 255) / 256, 256, 0, stream>>>(src, dst, iso, isi);
    degree_finalize<<<(N_NODES + 255) / 256, 256, 0, stream>>>(iso, isi);

    const int MB80 = N_NODES / 80;   // 625 blocks, exact
    const int MB16 = N_NODES / 16;   // 3125 blocks, exact

    // 2) residual: res = inputs @ Wr + br
    gemm_wmma_f32<DIM, 80><<<MB80, 256, 0, stream>>>(inputs, Wr, nullptr, br, bufr, 0);

    // 3) four graph-conv layers
    const float* Wl[4] = {W1, W2, W3, W4};
    const float* bl[4] = {b1, b2, b3, b4};
    const float* x = inputs;
    for (int l = 0; l < 4; ++l) {
        // h = (x * iso) @ W
        gemm_wmma_f32<DIM, 80><<<MB80, 256, 0, stream>>>(x, Wl[l], iso, nullptr, bufh, 0);
        // agg = segment_sum(h[src] * ew, dst)
        zero_f32<<<(N_NODES * DIM + 255) / 256, 256, 0, stream>>>(bufa, N_NODES * DIM);
        edge_scatter<<<N_EDGES / 8, 256, 0, stream>>>(src, dst, ew, bufh, bufa);
        // x = relu(agg * isi + b [+ res on layer 4])
        conv_epilogue<<<(N_NODES * (DIM / 4) + 255) / 256, 256, 0, stream>>>(
            (const float4*)bufa, isi, (const float4*)bl[l],
            (l == 3) ? (const float4*)bufr : nullptr, (float4*)bufx, 1);
        x = bufx;
    }

    // 4) out = x @ Wo + bo   (NOUT = 16 -> one wave per block, 16-row strips)
    gemm_wmma_f32<NCLS, 16><<<MB16, 32, 0, stream>>>(bufx, Wo, nullptr, bo, (float*)d_out, 0);
}